// GraphNeuralNetwork_18554258719469
// MI455X (gfx1250) — compile-verified
//
#include <hip/hip_runtime.h>
#include <cmath>

typedef __attribute__((ext_vector_type(2))) float v2f;
typedef __attribute__((ext_vector_type(8))) float v8f;

// ---------------------------------------------------------------------------
// Degree / normalization kernels
// ---------------------------------------------------------------------------
__global__ void deg_init_kernel(float* __restrict__ deg, int n) {
    int i = blockIdx.x * blockDim.x + threadIdx.x;
    if (i < n) deg[i] = 1.0f;  // self-loop
}

__global__ void deg_accum_kernel(float* __restrict__ deg,
                                 const long long* __restrict__ dst, int e) {
    int i = blockIdx.x * blockDim.x + threadIdx.x;
    if (i < e) unsafeAtomicAdd(&deg[(int)dst[i]], 1.0f);
}

__global__ void deg_rsqrt_kernel(float* __restrict__ deg, int n) {
    int i = blockIdx.x * blockDim.x + threadIdx.x;
    if (i < n) deg[i] = rsqrtf(deg[i]);  // deg >= 1 always (self-loops)
}

// ---------------------------------------------------------------------------
// Dense GEMM via FP32 WMMA: H[N,KOUT] = X[N,64] @ W[64,KOUT]
// Block = 128 threads = 4 waves; each wave computes one 16-row x KOUT tile.
// Grid = N/(16*4) blocks exactly (N=80000 -> 1250), so EXEC is all-ones for
// every WMMA (required by ISA).
//
// FP32 WMMA fragment layout (ISA 7.12.2):
//   A (16x4, 2 VGPRs):  lane L: M = L&15, K = vgpr + 2*(L>=16)
//   B (4x16, 2 VGPRs):  lane L: N = L&15, K = vgpr + 2*(L>=16)
//   C/D (16x16, 8 VGPRs): vgpr r: M = r + 8*(L>=16), N = L&15
//
// W is staged in LDS pre-swizzled into fragment order so each lane's B
// fragment (rows k,k+1 at column n) is one contiguous 8-byte chunk:
//   Wlds[(k>>1)*KOUT*2 + n*2 + (k&1)] = W[k][n]
// -> single ds_load_b64 per B fragment, no repack movs.
// ---------------------------------------------------------------------------
template <int KOUT>
__global__ __launch_bounds__(128) void gcn_gemm_wmma(
    const float* __restrict__ X, const float* __restrict__ W,
    float* __restrict__ H) {
    __shared__ float Wlds[64 * KOUT];
    for (int i = threadIdx.x; i < 64 * KOUT; i += 128) {
        int k = i / KOUT;
        int n = i - k * KOUT;
        Wlds[(k >> 1) * KOUT * 2 + n * 2 + (k & 1)] = W[i];
    }
    __syncthreads();

    const int wave = threadIdx.x >> 5;
    const int lane = threadIdx.x & 31;
    const int hi   = lane >> 4;   // 0 or 1 (which half-wave)
    const int l15  = lane & 15;
    const int row0 = (blockIdx.x * 4 + wave) * 16;

    const float* __restrict__ Xrow = X + (size_t)(row0 + l15) * 64;

    v8f acc[KOUT / 16];
#pragma unroll
    for (int nt = 0; nt < KOUT / 16; ++nt) acc[nt] = (v8f)0.0f;

#pragma unroll
    for (int k0 = 0; k0 < 64; k0 += 4) {
        const int ka = k0 + 2 * hi;   // even
        v2f a = *(const v2f*)(Xrow + ka);
#pragma unroll
        for (int nt = 0; nt < KOUT / 16; ++nt) {
            v2f b = *(const v2f*)(&Wlds[(ka >> 1) * KOUT * 2 + (nt * 16 + l15) * 2]);
            acc[nt] = __builtin_amdgcn_wmma_f32_16x16x4_f32(
                false, a, false, b, (short)0, acc[nt], false, false);
        }
    }

#pragma unroll
    for (int nt = 0; nt < KOUT / 16; ++nt) {
#pragma unroll
        for (int r = 0; r < 8; ++r) {
            H[(size_t)(row0 + r + 8 * hi) * KOUT + nt * 16 + l15] = acc[nt][r];
        }
    }
}

// ---------------------------------------------------------------------------
// out[i,f] = H[i,f]*dinv[i]^2 + b[f]   (self-loop message + bias, plain store)
// float4-vectorized: one thread per 4 features.
// ---------------------------------------------------------------------------
template <int K>
__global__ void self_bias_init_kernel(float4* __restrict__ out,
                                      const float4* __restrict__ H,
                                      const float* __restrict__ dinv,
                                      const float* __restrict__ b, int n) {
    int t = blockIdx.x * blockDim.x + threadIdx.x;   // over n*K/4
    if (t >= n * (K / 4)) return;
    int i  = t / (K / 4);
    int f4 = t - i * (K / 4);
    float di = dinv[i];
    float s  = di * di;
    float4 h = H[t];
    const float* bb = b + f4 * 4;
    float4 o;
    o.x = h.x * s + bb[0];
    o.y = h.y * s + bb[1];
    o.z = h.z * s + bb[2];
    o.w = h.w * s + bb[3];
    out[t] = o;
}

// ---------------------------------------------------------------------------
// Edge scatter: out[dst, f] += H[src, f] * dinv[src]*dinv[dst]
// K/4 threads per edge; each thread gathers a float4 of H (global_load_b128)
// and issues 4 global_atomic_add_f32.
// ---------------------------------------------------------------------------
template <int K>
__global__ void edge_scatter_kernel(float* __restrict__ out,
                                    const float* __restrict__ H,
                                    const float* __restrict__ dinv,
                                    const long long* __restrict__ src,
                                    const long long* __restrict__ dst, int e) {
    constexpr int TPE = K / 4;   // threads per edge
    unsigned int t    = blockIdx.x * blockDim.x + threadIdx.x;
    unsigned int edge = t / TPE;
    unsigned int f    = (t - edge * TPE) * 4;
    if (edge >= (unsigned int)e) return;
    int s = (int)src[edge];
    int d = (int)dst[edge];
    float norm = dinv[s] * dinv[d];
    float4 h = *(const float4*)(H + (size_t)s * K + f);
    float* o = out + (size_t)d * K + f;
    unsafeAtomicAdd(o + 0, h.x * norm);
    unsafeAtomicAdd(o + 1, h.y * norm);
    unsafeAtomicAdd(o + 2, h.z * norm);
    unsafeAtomicAdd(o + 3, h.w * norm);
}

// ---------------------------------------------------------------------------
// In-place ELU (alpha = 1): x > 0 ? x : expm1(x), float4-vectorized
// ---------------------------------------------------------------------------
__global__ void elu_kernel(float4* __restrict__ a, int n4) {
    int t = blockIdx.x * blockDim.x + threadIdx.x;
    if (t < n4) {
        float4 v = a[t];
        v.x = v.x > 0.0f ? v.x : expm1f(v.x);
        v.y = v.y > 0.0f ? v.y : expm1f(v.y);
        v.z = v.z > 0.0f ? v.z : expm1f(v.z);
        v.w = v.w > 0.0f ? v.w : expm1f(v.w);
        a[t] = v;
    }
}

// ---------------------------------------------------------------------------
// Launch
// ---------------------------------------------------------------------------
extern "C" void kernel_launch(void* const* d_in, const int* in_sizes, int n_in,
                              void* d_out, int out_size, void* d_ws,
                              size_t ws_size, hipStream_t stream) {
    const float* x  = (const float*)d_in[0];
    const float* W1 = (const float*)d_in[1];
    const float* b1 = (const float*)d_in[2];
    const float* W2 = (const float*)d_in[3];
    const float* b2 = (const float*)d_in[4];
    const float* W3 = (const float*)d_in[5];
    const float* b3 = (const float*)d_in[6];
    const long long* ei = (const long long*)d_in[7];

    const int N = in_sizes[0] / 64;      // 80000
    const int E = in_sizes[7] / 2;       // 1280000
    const long long* src = ei;
    const long long* dst = ei + E;

    // workspace layout: dinv[N] | H[N*64] | A[N*64]   (all 16B-aligned)
    float* dinv = (float*)d_ws;
    float* H    = dinv + N;
    float* A    = H + (size_t)N * 64;

    const int T = 256;

    // --- degree / normalization (shared by all layers) ---
    deg_init_kernel<<<(N + T - 1) / T, T, 0, stream>>>(dinv, N);
    deg_accum_kernel<<<(E + T - 1) / T, T, 0, stream>>>(dinv, dst, E);
    deg_rsqrt_kernel<<<(N + T - 1) / T, T, 0, stream>>>(dinv, N);

    const int gemm_blocks = N / 64;              // 16 rows/wave * 4 waves
    const int n4_64 = N * 64 / 4;
    const int n4_32 = N * 32 / 4;
    const unsigned sc64 = ((unsigned)E * 16 + T - 1) / T;   // K=64: 16 thr/edge
    const unsigned sc32 = ((unsigned)E * 8 + T - 1) / T;    // K=32:  8 thr/edge

    // --- layer 1: x -> A ---
    gcn_gemm_wmma<64><<<gemm_blocks, 128, 0, stream>>>(x, W1, H);
    self_bias_init_kernel<64><<<(n4_64 + T - 1) / T, T, 0, stream>>>(
        (float4*)A, (const float4*)H, dinv, b1, N);
    edge_scatter_kernel<64><<<sc64, T, 0, stream>>>(A, H, dinv, src, dst, E);
    elu_kernel<<<(n4_64 + T - 1) / T, T, 0, stream>>>((float4*)A, n4_64);

    // --- layer 2: A -> A (H is scratch; launches serialize on stream) ---
    gcn_gemm_wmma<64><<<gemm_blocks, 128, 0, stream>>>(A, W2, H);
    self_bias_init_kernel<64><<<(n4_64 + T - 1) / T, T, 0, stream>>>(
        (float4*)A, (const float4*)H, dinv, b2, N);
    edge_scatter_kernel<64><<<sc64, T, 0, stream>>>(A, H, dinv, src, dst, E);
    elu_kernel<<<(n4_64 + T - 1) / T, T, 0, stream>>>((float4*)A, n4_64);

    // --- layer 3: A -> d_out (KOUT = 32, no activation) ---
    float* out = (float*)d_out;
    gcn_gemm_wmma<32><<<gemm_blocks, 128, 0, stream>>>(A, W3, H);
    self_bias_init_kernel<32><<<(n4_32 + T - 1) / T, T, 0, stream>>>(
        (float4*)out, (const float4*)H, dinv, b3, N);
    edge_scatter_kernel<32><<<sc32, T, 0, stream>>>(out, H, dinv, src, dst, E);
}